// R_GNN_Model_78417512891196
// MI455X (gfx1250) — compile-verified
//
#include <hip/hip_runtime.h>
#include <hip/hip_bf16.h>

// ---------------------------------------------------------------------------
// RGCN (2-layer, basis-decomposed, mean aggregation) for MI455X / gfx1250.
// Dense per-relation projections via V_WMMA_F32_16X16X4_F32 (wave32 WMMA).
// Panel staging via GLOBAL_LOAD_ASYNC_TO_LDS_B128 when available (ASYNCcnt),
// edge gather/scatter via float4 loads + global_atomic_add_f32.
// ---------------------------------------------------------------------------

#define N_NODES 50000
#define N_EDGES 640000
#define IN_DIM  128
#define HID_DIM 128
#define OUT_DIM 64
#define N_REL   8
#define N_BASES 30

#define C1 (N_REL * HID_DIM + HID_DIM)   // 1152 columns (8 relations + root)
#define C2 (N_REL * OUT_DIM + OUT_DIM)   // 576 columns

typedef float v2f __attribute__((ext_vector_type(2)));
typedef float v8f __attribute__((ext_vector_type(8)));
typedef int   v4i __attribute__((vector_size(16)));   // matches builtin's pointee type

#if defined(__has_builtin)
#if __has_builtin(__builtin_amdgcn_global_load_async_to_lds_b128)
#define USE_ASYNC_LDS 1
#endif
#endif

// Copy 16 bytes global -> LDS. Async (no VGPR round-trip) when supported.
__device__ __forceinline__ void copy16_to_lds(const float* __restrict__ g,
                                              float* __restrict__ l) {
#ifdef USE_ASYNC_LDS
  __builtin_amdgcn_global_load_async_to_lds_b128(
      (__attribute__((address_space(1))) v4i*)g,
      (__attribute__((address_space(3))) v4i*)l, 0, 0);
#else
  *reinterpret_cast<float4*>(l) = *reinterpret_cast<const float4*>(g);
#endif
}

__device__ __forceinline__ void wait_stage_done() {
#ifdef USE_ASYNC_LDS
#if __has_builtin(__builtin_amdgcn_s_wait_asynccnt)
  __builtin_amdgcn_s_wait_asynccnt(0);
#else
  asm volatile("s_wait_asynccnt 0x0" ::: "memory");
#endif
#endif
  __syncthreads();
}

// ---------------------------------------------------------------------------
// Build K-transposed extended weights: WT[c][k], c in [0, R*outd) -> relation
// weights sum_b comp[r,b]*basis[b][k][o];  c in [R*outd, C) -> root[k][o].
// K-contiguous rows make the GEMM's B-operand an 8B contiguous LDS load.
// ---------------------------------------------------------------------------
__global__ void build_wext_t(const float* __restrict__ basis,
                             const float* __restrict__ comp,
                             const float* __restrict__ root,
                             float* __restrict__ WT, int K, int outd) {
  const int C = N_REL * outd + outd;
  const int idx = blockIdx.x * blockDim.x + threadIdx.x;
  if (idx >= K * C) return;
  const int c = idx / K;
  const int k = idx - c * K;
  float v;
  if (c < N_REL * outd) {
    const int r = c / outd;
    const int o = c - r * outd;
    float s = 0.0f;
#pragma unroll
    for (int b = 0; b < N_BASES; ++b)
      s += comp[r * N_BASES + b] * basis[((size_t)b * K + k) * outd + o];
    v = s;
  } else {
    v = root[(size_t)k * outd + (c - N_REL * outd)];
  }
  WT[(size_t)c * K + k] = v;
}

// ---------------------------------------------------------------------------
// Tiled fp32 WMMA GEMM: Y[M,C] = X[M,K] * WT[C,K]^T.
// Block tile 128(M) x 64(N), K panel of 32 staged in LDS (both operands
// K-contiguous). 8 waves; wave w owns rows [w*16, w*16+16), four N tiles.
// ---------------------------------------------------------------------------
#define BM 128
#define BN 64
#define BK 32
#define XS_STRIDE 36   // (36*r)%64 distinct for 16 consecutive r -> conflict-free
#define WT_STRIDE 36

__global__ __launch_bounds__(256, 2) void wmma_gemm_f32(
    const float* __restrict__ X, const float* __restrict__ WT,
    float* __restrict__ Y, int M, int K, int C) {
  __shared__ float Xs[BM * XS_STRIDE];
  __shared__ float Wt[BN * WT_STRIDE];

  const int tid  = threadIdx.x;
  const int wave = tid >> 5;
  const int lane = tid & 31;
  const int half = lane >> 4;   // selects K-pair (A/B) and M-offset (D)
  const int l15  = lane & 15;   // M row (A) / N col (B, D)
  const int m0 = blockIdx.y * BM;
  const int n0 = blockIdx.x * BN;

  v8f acc[4] = {};

  for (int k0 = 0; k0 < K; k0 += BK) {
    // ---- stage X panel [BM x BK]; clamp M edge (rows >= M only feed masked D rows)
#pragma unroll
    for (int it = 0; it < 4; ++it) {           // 1024 x 16B chunks / 256 thr
      const int idx = it * 256 + tid;
      const int r = idx >> 3, q = idx & 7;
      int gr = m0 + r;
      gr = gr < M ? gr : (M - 1);
      copy16_to_lds(&X[(size_t)gr * K + k0 + q * 4], &Xs[r * XS_STRIDE + q * 4]);
    }
    // ---- stage W panel [BN x BK] from K-transposed weights ----
#pragma unroll
    for (int it = 0; it < 2; ++it) {           // 512 x 16B chunks / 256 thr
      const int idx = it * 256 + tid;
      const int r = idx >> 3, q = idx & 7;
      copy16_to_lds(&WT[(size_t)(n0 + r) * K + k0 + q * 4], &Wt[r * WT_STRIDE + q * 4]);
    }
    // prefetch next K panel of WT into cache (global_prefetch_b8)
    if (k0 + BK < K)
      __builtin_prefetch(&WT[(size_t)(n0 + (tid >> 2)) * K + k0 + BK], 0, 1);
    wait_stage_done();

    // ---- compute: 8 K-steps of 4, four 16x16 N tiles per wave ----
#pragma unroll
    for (int kk = 0; kk < BK; kk += 4) {
      const int ka = kk + 2 * half;
      const v2f a = *reinterpret_cast<const v2f*>(&Xs[(wave * 16 + l15) * XS_STRIDE + ka]);
#pragma unroll
      for (int j = 0; j < 4; ++j) {
        const v2f b = *reinterpret_cast<const v2f*>(&Wt[(j * 16 + l15) * WT_STRIDE + ka]);
        acc[j] = __builtin_amdgcn_wmma_f32_16x16x4_f32(
            false, a, false, b, (short)0, acc[j], false, false);
      }
    }
    __syncthreads();
  }

  // ---- store D: VGPR v / lane-half -> (M = v + half*8), N = l15 ----
  const int grbase = m0 + wave * 16 + half * 8;
#pragma unroll
  for (int j = 0; j < 4; ++j) {
#pragma unroll
    for (int v = 0; v < 8; ++v) {
      const int gr = grbase + v;
      if (gr < M) Y[(size_t)gr * C + n0 + j * 16 + l15] = acc[j][v];
    }
  }
}

// ---------------------------------------------------------------------------
// Edge-phase helpers
// ---------------------------------------------------------------------------
__global__ void zero_f32(float* __restrict__ p, long long n) {
  long long i = (long long)blockIdx.x * blockDim.x + threadIdx.x;
  if (i < n) p[i] = 0.0f;
}

__global__ void count_edges(const int* __restrict__ ei, const int* __restrict__ et,
                            float* __restrict__ cnt) {
  int e = blockIdx.x * blockDim.x + threadIdx.x;
  if (e >= N_EDGES) return;
  const int d = ei[N_EDGES + e];
  const int r = et[e];
  atomicAdd(&cnt[(size_t)d * N_REL + r], 1.0f);
}

__global__ void inv_counts(float* __restrict__ cnt, long long n) {
  long long i = (long long)blockIdx.x * blockDim.x + threadIdx.x;
  if (i >= n) return;
  const float c = cnt[i];
  cnt[i] = (c > 0.0f) ? (1.0f / c) : 0.0f;
}

// outd/4 threads cooperate per edge; each handles one float4 slice.
__global__ void scatter_mean(const float* __restrict__ XP,
                             const int* __restrict__ ei, const int* __restrict__ et,
                             const float* __restrict__ inv,
                             float* __restrict__ AGG, int C, int outd) {
  const int tpe = outd >> 2;
  const long long gid = (long long)blockIdx.x * blockDim.x + threadIdx.x;
  const long long e = gid / tpe;
  const int l = (int)(gid - e * tpe);
  if (e >= N_EDGES) return;
  const int s = ei[e];
  const int d = ei[N_EDGES + e];
  const int r = et[e];
  const float w = inv[(size_t)d * N_REL + r];
  const float4 v = *reinterpret_cast<const float4*>(&XP[(size_t)s * C + r * outd + l * 4]);
  float* dst = &AGG[(size_t)d * outd + l * 4];
  atomicAdd(dst + 0, v.x * w);
  atomicAdd(dst + 1, v.y * w);
  atomicAdd(dst + 2, v.z * w);
  atomicAdd(dst + 3, v.w * w);
}

// out[n,o] = relu(AGG[n,o] + XP[n, R*outd + o] + bias[o])
__global__ void finalize_relu(const float* __restrict__ AGG, const float* __restrict__ XP,
                              const float* __restrict__ bias, float* __restrict__ out,
                              int C, int outd) {
  const long long i = (long long)blockIdx.x * blockDim.x + threadIdx.x;
  if (i >= (long long)N_NODES * outd) return;
  const long long n = i / outd;
  const int o = (int)(i - n * outd);
  const float v = AGG[i] + XP[(size_t)n * C + N_REL * outd + o] + bias[o];
  out[i] = v > 0.0f ? v : 0.0f;
}

// ---------------------------------------------------------------------------
// Launch
// ---------------------------------------------------------------------------
extern "C" void kernel_launch(void* const* d_in, const int* in_sizes, int n_in,
                              void* d_out, int out_size, void* d_ws, size_t ws_size,
                              hipStream_t stream) {
  const float* x      = (const float*)d_in[0];
  const int*   ei     = (const int*)d_in[1];   // [2, E]
  const int*   et     = (const int*)d_in[2];   // [E]
  const float* basis1 = (const float*)d_in[3];
  const float* comp1  = (const float*)d_in[4];
  const float* root1  = (const float*)d_in[5];
  const float* bias1  = (const float*)d_in[6];
  const float* basis2 = (const float*)d_in[7];
  const float* comp2  = (const float*)d_in[8];
  const float* root2  = (const float*)d_in[9];
  const float* bias2  = (const float*)d_in[10];
  float* out = (float*)d_out;

  // ---- workspace carve-up (all float, 16B-aligned offsets) ----
  float* p = (float*)d_ws;
  float* WT1  = p; p += (size_t)IN_DIM * C1;            // 147456  (stored [C1][K])
  float* WT2  = p; p += (size_t)HID_DIM * C2;           // 73728   (stored [C2][K])
  float* cnt  = p; p += (size_t)N_NODES * N_REL;        // 400000
  float* AGG1 = p; p += (size_t)N_NODES * HID_DIM;      // 6.4M
  float* H    = p; p += (size_t)N_NODES * HID_DIM;      // 6.4M
  float* AGG2 = p; p += (size_t)N_NODES * OUT_DIM;      // 3.2M
  float* XP   = p; p += (size_t)N_NODES * C1;           // 57.6M (reused for layer 2)

  const int T = 256;
  auto blks = [](long long n, int t) { return (unsigned)((n + t - 1) / t); };

  // ---- weights (K-transposed, root folded in as extra columns) ----
  build_wext_t<<<blks((long long)IN_DIM * C1, T), T, 0, stream>>>(basis1, comp1, root1, WT1, IN_DIM, HID_DIM);
  build_wext_t<<<blks((long long)HID_DIM * C2, T), T, 0, stream>>>(basis2, comp2, root2, WT2, HID_DIM, OUT_DIM);

  // ---- (dst, rel) mean normalizers (shared by both layers) ----
  zero_f32<<<blks((long long)N_NODES * N_REL, T), T, 0, stream>>>(cnt, (long long)N_NODES * N_REL);
  count_edges<<<blks(N_EDGES, T), T, 0, stream>>>(ei, et, cnt);
  inv_counts<<<blks((long long)N_NODES * N_REL, T), T, 0, stream>>>(cnt, (long long)N_NODES * N_REL);

  // ---- layer 1 ----
  {
    dim3 g(C1 / BN, (N_NODES + BM - 1) / BM);
    wmma_gemm_f32<<<g, T, 0, stream>>>(x, WT1, XP, N_NODES, IN_DIM, C1);
    zero_f32<<<blks((long long)N_NODES * HID_DIM, T), T, 0, stream>>>(AGG1, (long long)N_NODES * HID_DIM);
    scatter_mean<<<blks((long long)N_EDGES * (HID_DIM / 4), T), T, 0, stream>>>(XP, ei, et, cnt, AGG1, C1, HID_DIM);
    finalize_relu<<<blks((long long)N_NODES * HID_DIM, T), T, 0, stream>>>(AGG1, XP, bias1, H, C1, HID_DIM);
  }

  // ---- layer 2 (reuses XP region; stream ordering guarantees layer-1 reads done) ----
  {
    dim3 g(C2 / BN, (N_NODES + BM - 1) / BM);
    wmma_gemm_f32<<<g, T, 0, stream>>>(H, WT2, XP, N_NODES, HID_DIM, C2);
    zero_f32<<<blks((long long)N_NODES * OUT_DIM, T), T, 0, stream>>>(AGG2, (long long)N_NODES * OUT_DIM);
    scatter_mean<<<blks((long long)N_EDGES * (OUT_DIM / 4), T), T, 0, stream>>>(XP, ei, et, cnt, AGG2, C2, OUT_DIM);
    finalize_relu<<<blks((long long)N_NODES * OUT_DIM, T), T, 0, stream>>>(AGG2, XP, bias2, out, C2, OUT_DIM);
  }
}